// Autoencoder_62457414418559
// MI455X (gfx1250) — compile-verified
//
#include <hip/hip_runtime.h>

// ---------------------------------------------------------------------------
// TopK sparse autoencoder forward for MI455X (gfx1250, wave32).
// Memory-bound (~3.3 GB traffic @ 23.3 TB/s ~ 140us; f32 WMMA compute ~7us),
// so use V_WMMA_F32_16X16X4_F32 (exact f32 semantics) and keep HBM streaming
// overlapped with compute via TDM (tensor_load_to_lds) double buffering.
// ---------------------------------------------------------------------------

typedef float v2f __attribute__((ext_vector_type(2)));
typedef float v8f __attribute__((ext_vector_type(8)));
typedef unsigned uint32x4 __attribute__((ext_vector_type(4)));
typedef int int32x8 __attribute__((ext_vector_type(8)));
typedef int int32x4 __attribute__((ext_vector_type(4)));

#define BM 128
#define BN 128
#define BK 32
#define LDSS 36  // LDS row stride (floats): 16B-aligned rows, conflict-free frags

// LDS byte offset of a __shared__ pointer: flat LDS address low 32 bits are the
// LDS offset (aperture lives in the high bits).
__device__ __forceinline__ unsigned lds_off(const void* p) {
  return (unsigned)(unsigned long long)p;
}

// TDM: DMA one 128 x 32-float tile (global row stride = k_stride elems) into
// LDS with hardware padding: after every 32 DWORDs insert 4 DWORDs -> LDS row
// stride 36 floats, matching the compute-side layout. Issued once per wave.
__device__ __forceinline__ void tdm_load_tile(unsigned lds_addr,
                                              const float* gaddr,
                                              unsigned k_stride) {
  unsigned long long ga = (unsigned long long)gaddr;
  uint32x4 g0;
  g0[0] = 1u;                                 // count=1 (valid descriptor)
  g0[1] = lds_addr;                           // lds_addr
  g0[2] = (unsigned)ga;                       // global_addr[31:0]
  g0[3] = (unsigned)(ga >> 32) | (2u << 30);  // global_addr[56:32] | type=2
  int32x8 g1;
  // data_size=4B(2), pad_enable, pad_interval=32 DW(code 4), pad_amount=4 DW(code 3)
  g1[0] = (int)((2u << 16) | (1u << 20) | (4u << 22) | (3u << 25));
  g1[1] = (int)((k_stride & 0xFFFFu) << 16);       // tensor_dim0[15:0]
  g1[2] = (int)((k_stride >> 16) | (128u << 16));  // tensor_dim0[31:16] | tensor_dim1[15:0]=128
  g1[3] = (int)((unsigned)BK << 16);               // tensor_dim1[31:16]=0 | tile_dim0=32
  g1[4] = (int)128u;                               // tile_dim1=128 | tile_dim2=0
  g1[5] = (int)k_stride;                           // tensor_dim0_stride[31:0]
  g1[6] = 0;                                       // stride[47:32] | dim1_stride lo
  g1[7] = 0;
  int32x4 z4 = {0, 0, 0, 0};                       // groups 2/3 unused (2D tile)
  int32x8 z8 = {0, 0, 0, 0, 0, 0, 0, 0};
  __builtin_amdgcn_tensor_load_to_lds(g0, g1, z4, z4, z8, 0);
}

// C[m,n] = sum_k (A[m,k] - kbias[k]) * Bm[n,k]  + nbias[n]
// A: [M,K] row-major, Bm: [N,K] row-major (both K-contiguous -> "NT" GEMM).
template <bool SUB_KBIAS>
__global__ __launch_bounds__(256) void gemm_nt_f32_wmma(
    const float* __restrict__ A, const float* __restrict__ Bm,
    const float* __restrict__ kbias, const float* __restrict__ nbias,
    float* __restrict__ C, int M, int N, int K) {
  __shared__ float sA[2][BM * LDSS];
  __shared__ float sB[2][BN * LDSS];
  __shared__ float sPB[2][BK];

  const int tid  = threadIdx.x;
  const int lane = tid & 31;
  const int wave = tid >> 5;    // 8 waves
  const int waveM = wave & 3;   // 4 waves x 32 rows
  const int waveN = wave >> 2;  // 2 waves x 64 cols

  const int m0 = blockIdx.x * BM;  // M-tile fastest -> weight panel L2 reuse
  const int n0 = blockIdx.y * BN;
  const float* Atile = A + (size_t)m0 * K;
  const float* Btile = Bm + (size_t)n0 * K;

  v8f acc[2][4];
  {
    v8f zero = {};
#pragma unroll
    for (int i = 0; i < 2; ++i)
#pragma unroll
      for (int j = 0; j < 4; ++j) acc[i][j] = zero;
  }

  // Prologue: wave 0 DMAs the first tiles, waits TENSORcnt, barrier publishes.
  if (tid < 32) {
    tdm_load_tile(lds_off(&sA[0][0]), Atile, (unsigned)K);
    tdm_load_tile(lds_off(&sB[0][0]), Btile, (unsigned)K);
    if (SUB_KBIAS) sPB[0][lane] = kbias[lane];
    __builtin_amdgcn_s_wait_tensorcnt(0);
  }
  __syncthreads();

  int buf = 0;
  for (int k0 = 0; k0 < K; k0 += BK) {
    const int k1 = k0 + BK;
    // Prefetch next chunk into the other buffer while we compute (TDM is async).
    if (k1 < K && tid < 32) {
      tdm_load_tile(lds_off(&sA[buf ^ 1][0]), Atile + k1, (unsigned)K);
      tdm_load_tile(lds_off(&sB[buf ^ 1][0]), Btile + k1, (unsigned)K);
      if (SUB_KBIAS) sPB[buf ^ 1][lane] = kbias[k1 + lane];
    }

#pragma unroll
    for (int kk = 0; kk < BK; kk += 4) {
      // A frag 16x4: lanes 0-15 hold M rows (K=0,1), lanes 16-31 (K=2,3).
      const int colpair = kk + (lane >> 4) * 2;
      v2f aF[2], bF[4];
      v2f pb;
      if (SUB_KBIAS) pb = *(const v2f*)&sPB[buf][colpair];
#pragma unroll
      for (int mi = 0; mi < 2; ++mi) {
        int r = waveM * 32 + mi * 16 + (lane & 15);
        aF[mi] = *(const v2f*)&sA[buf][r * LDSS + colpair];
        if (SUB_KBIAS) aF[mi] = aF[mi] - pb;
      }
#pragma unroll
      for (int ni = 0; ni < 4; ++ni) {
        int r = waveN * 64 + ni * 16 + (lane & 15);
        bF[ni] = *(const v2f*)&sB[buf][r * LDSS + colpair];
      }
#pragma unroll
      for (int mi = 0; mi < 2; ++mi)
#pragma unroll
        for (int ni = 0; ni < 4; ++ni)
          acc[mi][ni] = __builtin_amdgcn_wmma_f32_16x16x4_f32(
              false, aF[mi], false, bF[ni], (short)0, acc[mi][ni], false, false);
    }

    if (tid < 32) __builtin_amdgcn_s_wait_tensorcnt(0);
    __syncthreads();
    buf ^= 1;
  }

  // Epilogue: C/D layout -> VGPR r: M = r + (lane/16)*8, N = lane%16.
#pragma unroll
  for (int mi = 0; mi < 2; ++mi)
#pragma unroll
    for (int ni = 0; ni < 4; ++ni) {
      int col = n0 + waveN * 64 + ni * 16 + (lane & 15);
      float bias = nbias[col];
#pragma unroll
      for (int r = 0; r < 8; ++r) {
        int row = m0 + waveM * 32 + mi * 16 + ((lane >> 4) << 3) + r;
        C[(size_t)row * N + col] = acc[mi][ni][r] + bias;
      }
    }
}

// ---------------------------------------------------------------------------
// Per-row top-k (k=64 of 32768) via 3-level radix select on monotone uint keys,
// then ReLU + dense scatter. One block per row; row re-reads hit L2.
// ---------------------------------------------------------------------------
__device__ __forceinline__ unsigned fkey(float f) {
  unsigned u = __float_as_uint(f);
  return (u & 0x80000000u) ? ~u : (u | 0x80000000u);
}

__global__ __launch_bounds__(256) void topk_relu_scatter(
    const float* __restrict__ pre, float* __restrict__ out,
    const int* __restrict__ kp, int L) {
  __shared__ unsigned hist[4096];
  __shared__ unsigned s_b, s_rem, s_tie;
  const int tid = threadIdx.x;
  const unsigned k = (unsigned)kp[0];
  const float* x = pre + (size_t)blockIdx.x * L;
  float* y = out + (size_t)blockIdx.x * L;

  // pass 1: top 12 bits
  for (int i = tid; i < 4096; i += 256) hist[i] = 0;
  __syncthreads();
  for (int i = tid; i < L; i += 256) atomicAdd(&hist[fkey(x[i]) >> 20], 1u);
  __syncthreads();
  if (tid == 0) {
    unsigned c = 0; int b = 4095;
    for (; b > 0; --b) { if (c + hist[b] >= k) break; c += hist[b]; }
    s_b = (unsigned)b; s_rem = k - c;
  }
  __syncthreads();
  const unsigned b1 = s_b;

  // pass 2: bits 19..10 within bucket b1
  for (int i = tid; i < 1024; i += 256) hist[i] = 0;
  __syncthreads();
  for (int i = tid; i < L; i += 256) {
    unsigned key = fkey(x[i]);
    if ((key >> 20) == b1) atomicAdd(&hist[(key >> 10) & 1023u], 1u);
  }
  __syncthreads();
  if (tid == 0) {
    unsigned c = 0; int b = 1023;
    for (; b > 0; --b) { if (c + hist[b] >= s_rem) break; c += hist[b]; }
    s_b = (unsigned)b; s_rem -= c;
  }
  __syncthreads();
  const unsigned b2 = s_b;

  // pass 3: low 10 bits
  for (int i = tid; i < 1024; i += 256) hist[i] = 0;
  __syncthreads();
  for (int i = tid; i < L; i += 256) {
    unsigned key = fkey(x[i]);
    if ((key >> 10) == ((b1 << 10) | b2)) atomicAdd(&hist[key & 1023u], 1u);
  }
  __syncthreads();
  if (tid == 0) {
    unsigned c = 0; int b = 1023;
    for (; b > 0; --b) { if (c + hist[b] >= s_rem) break; c += hist[b]; }
    s_b = (unsigned)b; s_rem -= c;  // #threshold-equal elements to keep
    s_tie = 0;
  }
  __syncthreads();
  const unsigned keyThr = (b1 << 20) | (b2 << 10) | s_b;
  const unsigned rem = s_rem;

  // output: relu(top-k values) scattered into dense zeros
  for (int i = tid; i < L; i += 256) {
    float v = x[i];
    unsigned key = fkey(v);
    float o = 0.0f;
    if (key > keyThr) {
      o = v > 0.0f ? v : 0.0f;
    } else if (key == keyThr) {
      if (atomicAdd(&s_tie, 1u) < rem) o = v > 0.0f ? v : 0.0f;
    }
    y[i] = o;
  }
}

// ---------------------------------------------------------------------------
extern "C" void kernel_launch(void* const* d_in, const int* in_sizes, int n_in,
                              void* d_out, int out_size, void* d_ws,
                              size_t ws_size, hipStream_t stream) {
  (void)in_sizes; (void)n_in; (void)out_size; (void)d_ws; (void)ws_size;
  const float* x           = (const float*)d_in[0];  // [B,D]
  const float* W_enc       = (const float*)d_in[1];  // [L,D]
  const float* latent_bias = (const float*)d_in[2];  // [L]
  const float* pre_bias    = (const float*)d_in[3];  // [D]
  const float* W_dec       = (const float*)d_in[4];  // [D,L]
  const int*   kp          = (const int*)d_in[5];    // scalar k

  const int B = 4096, D = 4096, L = 32768;
  float* pre = (float*)d_out;        // latents_pre_act [B,L]
  float* lat = pre + (size_t)B * L;  // latents         [B,L]
  float* rec = lat + (size_t)B * L;  // recons          [B,D]

  dim3 blk(256);
  // encode: (x - pre_bias) @ W_enc^T + latent_bias
  gemm_nt_f32_wmma<true><<<dim3(B / BM, L / BN), blk, 0, stream>>>(
      x, W_enc, pre_bias, latent_bias, pre, B, L, D);
  // top-k + relu + scatter
  topk_relu_scatter<<<dim3(B), blk, 0, stream>>>(pre, lat, kp, L);
  // decode: latents @ W_dec^T + pre_bias (dense: latents must be written anyway)
  gemm_nt_f32_wmma<false><<<dim3(B / BM, D / BN), blk, 0, stream>>>(
      lat, W_dec, nullptr, pre_bias, rec, B, D, L);
}